// CrossEntropyLossWithGaussianSmoothedLabels_27075473834496
// MI455X (gfx1250) — compile-verified
//
#include <hip/hip_runtime.h>

#define NUM_CLASSES   722
#define ROWS_TOTAL    (16 * 4096)          // 65536
#define ROWS_PER_BLK  8                    // 8 waves of 32 per block
#define NUM_BLOCKS    (ROWS_TOTAL / ROWS_PER_BLK)   // 8192
#define CHUNKS        180                  // 722 = 180*4 + 2

typedef __attribute__((ext_vector_type(2))) float v2f;
typedef __attribute__((ext_vector_type(8))) float v8f;

__device__ __forceinline__ float wave_sum(float v) {
#pragma unroll
  for (int m = 16; m >= 1; m >>= 1) v += __shfl_xor(v, m, 32);
  return v;
}
__device__ __forceinline__ float wave_max(float v) {
#pragma unroll
  for (int m = 16; m >= 1; m >>= 1) v = fmaxf(v, __shfl_xor(v, m, 32));
  return v;
}

// One wave per row: single HBM pass over pred (float4 = GLOBAL_LOAD_B128),
// row cached in VGPRs between the max pass and the exp-sum pass.
__global__ __launch_bounds__(256)
void ce_gauss_rows(const float* __restrict__ pred,
                   const long long* __restrict__ target,
                   float* __restrict__ partial) {
  const int lane = threadIdx.x & 31;
  const int wave = threadIdx.x >> 5;
  const int row  = blockIdx.x * ROWS_PER_BLK + wave;
  const float* rp = pred + (size_t)row * NUM_CLASSES;

  // speculative prefetch of the back half of the row (global_prefetch_b8)
  __builtin_prefetch(rp + 512, 0, 0);

  float4 v[6];
  float m = -3.402823466e38f;
#pragma unroll
  for (int it = 0; it < 6; ++it) {
    const int chunk = it * 32 + lane;
    if (chunk < CHUNKS) {
      v[it] = ((const float4*)rp)[chunk];
      m = fmaxf(m, fmaxf(fmaxf(v[it].x, v[it].y), fmaxf(v[it].z, v[it].w)));
    }
  }
  const float tail = (lane < 2) ? rp[720 + lane] : -3.402823466e38f;
  m = fmaxf(m, tail);
  m = wave_max(m);

  float s = 0.0f;
#pragma unroll
  for (int it = 0; it < 6; ++it) {
    const int chunk = it * 32 + lane;
    if (chunk < CHUNKS) {
      s += __expf(v[it].x - m) + __expf(v[it].y - m) +
           __expf(v[it].z - m) + __expf(v[it].w - m);
    }
  }
  if (lane < 2) s += __expf(tail - m);
  s = wave_sum(s);
  const float lse = m + __logf(s);

  // Smoothed-label weights: replicate exact scatter-overwrite order.
  // Lanes 0..6 own classes t-3 .. t+3 (clipped writes land inside this window).
  const int t = (int)target[row];
  float wdot = 0.0f, wsum = 0.0f;
  if (lane < 7) {
    const int c = t - 3 + lane;
    if (c >= 0 && c < NUM_CLASSES) {
      // g(d) = exp(-2^d / 4), d = 0..3
      const float GV[4] = {0.7788007830714049f, 0.6065306597126334f,
                           0.3678794411714423f, 0.1353352832366127f};
      float w = 0.0f;
#pragma unroll
      for (int d = 3; d >= 0; --d) {
        int lo = t - d; if (lo < 0) lo = 0;
        int hi = t + d; if (hi > NUM_CLASSES - 1) hi = NUM_CLASSES - 1;
        if (c == lo) w = GV[d];
        if (c == hi) w = GV[d];
      }
      if (c == t) w = 1.0f;        // final scatter of exactly 1.0 at target
      wdot = w * rp[c];
      wsum = w;
    }
  }
  wdot = wave_sum(wdot);
  wsum = wave_sum(wsum);

  __shared__ float ls[ROWS_PER_BLK];
  if (lane == 0) ls[wave] = wsum * lse - wdot;
  __syncthreads();
  if (threadIdx.x == 0) {
    float acc = 0.0f;
#pragma unroll
    for (int i = 0; i < ROWS_PER_BLK; ++i) acc += ls[i];
    partial[blockIdx.x] = acc * (1.0f / (float)ROWS_TOTAL);
  }
}

// Single-wave deterministic 8192 -> 1 reduction via V_WMMA_F32_16X16X4_F32:
// A = 64 partials per iteration (16x4, 2 VGPRs/lane), B = ones (4x16),
// chained f32 accumulator. D[m,n] = sum over iters of row-sums of A, every
// column identical; grand total = per-lane component sum + shfl_xor(16).
__global__ __launch_bounds__(32)
void reduce_partials_wmma(const float* __restrict__ partial,
                          float* __restrict__ out) {
  const int lane = threadIdx.x;   // full wave active: EXEC all ones for WMMA
  v8f acc = {};
  v2f ones; ones[0] = 1.0f; ones[1] = 1.0f;
  for (int it = 0; it < NUM_BLOCKS / 64; ++it) {
    v2f a = *(const v2f*)(partial + it * 64 + lane * 2);
    acc = __builtin_amdgcn_wmma_f32_16x16x4_f32(
        /*neg_a=*/false, a, /*neg_b=*/false, ones,
        /*c_mod=*/(short)0, acc, /*reuse_a=*/false, /*reuse_b=*/false);
  }
  float s = acc[0] + acc[1] + acc[2] + acc[3] +
            acc[4] + acc[5] + acc[6] + acc[7];
  s += __shfl_xor(s, 16, 32);
  if (lane == 0) out[0] = s;
}

extern "C" void kernel_launch(void* const* d_in, const int* in_sizes, int n_in,
                              void* d_out, int out_size, void* d_ws, size_t ws_size,
                              hipStream_t stream) {
  const float*     pred   = (const float*)d_in[0];
  const long long* target = (const long long*)d_in[1];
  float* partial = (float*)d_ws;          // 8192 floats = 32 KB scratch

  ce_gauss_rows<<<NUM_BLOCKS, 256, 0, stream>>>(pred, target, partial);
  reduce_partials_wmma<<<1, 32, 0, stream>>>(partial, (float*)d_out);
}